// ItemEncoder_70755291234310
// MI455X (gfx1250) — compile-verified
//
#include <hip/hip_runtime.h>
#include <hip/hip_bf16.h>

typedef float v2f __attribute__((ext_vector_type(2)));
typedef float v8f __attribute__((ext_vector_type(8)));

#define INPUT_DIM 2000
#define OUT_DIM 8
#define TOKENS_PER_GROUP 16
#define WAVES_PER_BLOCK 4
#define K_PER_WAVE (INPUT_DIM / WAVES_PER_BLOCK)   // 500 -> 125 WMMA chunks/wave

// Zero-filled padding row for A-matrix rows 8..15 (W has only 8 rows).
// .bss device globals are zero-initialized at module load and never written,
// so this is deterministic across graph replays. Stays hot in L0/L2.
__device__ float ZROW[2048] = {};

// One block = 4 wave32s, one 16-token group.
// D = A(W padded 16x2000) x B(gathered rows^T 2000x16), K split across waves.
__global__ __launch_bounds__(32 * WAVES_PER_BLOCK)
void ItemEncoder_70755291234310_kernel(const int* __restrict__ inp,      // [B*L]
                                       const float* __restrict__ urm,    // [NTOKEN][2000]
                                       const float* __restrict__ W,      // [8][2000]
                                       const float* __restrict__ bias,   // [8]
                                       float* __restrict__ out)          // [B*L][8]
{
    __shared__ float red[WAVES_PER_BLOCK][32][OUT_DIM];   // 4 KB partial accumulators

    const int lane = threadIdx.x & 31;
    const int wave = threadIdx.x >> 5;
    const int n    = lane & 15;        // A row (M) / B column (token in group)
    const int half = lane >> 4;        // 0: holds K,K+1   1: holds K+2,K+3
    const int grp  = blockIdx.x;

    // Token for this lane's N column; row pointer into the 800 MB table (HBM stream).
    const int tok = inp[grp * TOKENS_PER_GROUP + n];
    const float* __restrict__ rowp = urm + (size_t)tok * INPUT_DIM;

    // Branch-free A row select, done ONCE: real W row for m<8, zero row otherwise.
    const float* __restrict__ arow = (n < OUT_DIM) ? (W + (size_t)n * INPUT_DIM) : ZROW;

    const int k0 = wave * K_PER_WAVE;
    const int k1 = k0 + K_PER_WAVE;
    const int half2 = half * 2;

    v8f c = {};

    // 125 chunks = 25 x 5: no remainder epilogue; 10 b64 loads in flight per window.
#pragma unroll 5
    for (int k = k0; k < k1; k += 4) {
        const int kb = k + half2;

        // B fragment: B[kb..kb+1][n] = gathered_row_n[kb..kb+1]   (HBM stream)
        v2f bmat = *(const v2f*)(rowp + kb);
        // A fragment: Wpad[m][kb..kb+1]                            (cache-resident)
        v2f amat = *(const v2f*)(arow + kb);

        // 16x16x4 fp32 WMMA accumulate: (neg_a, A, neg_b, B, c_mod, C, reuse_a, reuse_b)
        c = __builtin_amdgcn_wmma_f32_16x16x4_f32(false, amat, false, bmat,
                                                  (short)0, c, false, false);
    }

    // Cross-wave K reduction through LDS.
#pragma unroll
    for (int j = 0; j < OUT_DIM; ++j) red[wave][lane][j] = c[j];
    __syncthreads();

    // D layout: lane L<16 holds D[M=j][N=L] in c[j]  ->  out[tok_L][j].
    if (wave == 0 && lane < 16) {
        float acc[OUT_DIM];
#pragma unroll
        for (int j = 0; j < OUT_DIM; ++j) {
            float s = 0.0f;
#pragma unroll
            for (int w = 0; w < WAVES_PER_BLOCK; ++w) s += red[w][lane][j];
            acc[j] = s + bias[j];
        }
        float* op = out + (size_t)(grp * TOKENS_PER_GROUP + lane) * OUT_DIM;
        *(float4*)(op)     = make_float4(acc[0], acc[1], acc[2], acc[3]);
        *(float4*)(op + 4) = make_float4(acc[4], acc[5], acc[6], acc[7]);
    }
}

extern "C" void kernel_launch(void* const* d_in, const int* in_sizes, int n_in,
                              void* d_out, int out_size, void* d_ws, size_t ws_size,
                              hipStream_t stream) {
    (void)in_sizes; (void)n_in; (void)d_ws; (void)ws_size;

    const int*   inp  = (const int*)d_in[0];     // [64*200] int32 token ids
    const float* urm  = (const float*)d_in[1];   // [100000][2000] f32
    const float* W    = (const float*)d_in[2];   // [8][2000] f32
    const float* bias = (const float*)d_in[3];   // [8] f32
    float*       out  = (float*)d_out;           // [64*200][8] f32

    const int n_tokens = out_size / OUT_DIM;                 // 12800
    const int n_groups = n_tokens / TOKENS_PER_GROUP;        // 800

    ItemEncoder_70755291234310_kernel<<<n_groups, 32 * WAVES_PER_BLOCK, 0, stream>>>(
        inp, urm, W, bias, out);
}